// Model_1649267442008
// MI455X (gfx1250) — compile-verified
//
// Spherical CNN (S2Conv -> SO3Conv -> SO3Conv -> head) for MI455X (gfx1250).
//
// Every contraction (DFT, Legendre analysis, kernel projection, Z=xh*conj(yh),
// synthesis, inverse DFT) maps to ONE generic batched strided planar-complex
// GEMM engine running on V_WMMA_F32_16X16X4_F32 (fp32 parity with reference;
// ~45 GFLOP total, traffic-bound at 23.3 TB/s so fp32 matrix rate is ample).
// Complex modes are template params -> straight-line K loop: unconditional
// clamped loads + v_cndmask zeroing + 2..4 back-to-back v_wmma per K-step.
// Wigner-d / SOFT weights / twiddles are recomputed on device (fp64) per call.
// Workspace arena with reuse, peak ~201M floats (~805 MB).

#include <hip/hip_runtime.h>
#include <math.h>

typedef __attribute__((ext_vector_type(2))) float v2f;
typedef __attribute__((ext_vector_type(8))) float v8f;

#define PI_D 3.14159265358979323846

__device__ inline v8f wmma4(v2f a, v2f b, v8f c) {
  // D = A(16x4) * B(4x16) + C, fp32. (neg_a,A,neg_b,B,c_mod,C,reuse_a,reuse_b)
  return __builtin_amdgcn_wmma_f32_16x16x4_f32(false, a, false, b, (short)0, c, false, false);
}

// ---------------------------------------------------------------------------
// Batched strided planar-complex GEMM on WMMA f32 16x16x4.
//   C[z1,z2][m][n1,n2] = sum_k A[z1,z2][m][k] * (conjB? conj:id)(B[z1,z2][k][n1,n2])
// AC/BC: operand has imaginary plane. CC: store imaginary plane (else Re only).
// One wave owns a 16x16 C tile; 4 waves/block = 32x32 macro tile.
// WMMA at full EXEC; bounds handled by clamped loads + value selects.
// ---------------------------------------------------------------------------
template <bool AC, bool BC, bool CC>
__global__ __launch_bounds__(128)
void gemm_wmma_t(const float* __restrict__ Ar, const float* __restrict__ Ai,
                 const float* __restrict__ Br, const float* __restrict__ Bi,
                 float* __restrict__ Cr, float* __restrict__ Ci,
                 int M, int N1, int N2, int K, int nZ2,
                 long zA1, long zA2, long mA, long kA,
                 long zB1, long zB2, long kB, long nB1, long nB2,
                 long zC1, long zC2, long mC, long nC1, long nC2,
                 int conjB)
{
  const int lane = threadIdx.x & 31;
  const int wid  = threadIdx.x >> 5;
  const int half = lane >> 4;     // 0: lanes 0-15, 1: lanes 16-31
  const int l16  = lane & 15;
  const int N    = N1 * N2;
  const int tm0  = blockIdx.y * 32 + (wid >> 1) * 16;
  const int tn0  = blockIdx.x * 32 + (wid & 1) * 16;
  const int z    = blockIdx.z;
  const int z1   = z / nZ2, z2 = z - z1 * nZ2;

  // A fragment: lane half h holds (row = tm0+l16, k = k0+2h+{0,1})
  const int  aRow = tm0 + l16;
  const bool aOK  = aRow < M;
  const long aBase = z1 * zA1 + z2 * zA2 + (aOK ? (long)aRow * mA : 0); // clamped: row 0 if OOB

  // B fragment: lane half h holds (k = k0+2h+{0,1}, col = tn0+l16) [dual of C/D layout]
  const int  bCol = tn0 + l16;
  const bool bOK  = bCol < N;
  const int  bn1  = bOK ? bCol / N2 : 0;
  const int  bn2  = bOK ? bCol - bn1 * N2 : 0;
  const long bBase = z1 * zB1 + z2 * zB2 + (long)bn1 * nB1 + (long)bn2 * nB2;

  v8f crr = {0.f,0.f,0.f,0.f,0.f,0.f,0.f,0.f};
  v8f cri = {0.f,0.f,0.f,0.f,0.f,0.f,0.f,0.f};
  v8f cir = {0.f,0.f,0.f,0.f,0.f,0.f,0.f,0.f};
  v8f cii = {0.f,0.f,0.f,0.f,0.f,0.f,0.f,0.f};

  const int K4 = K & ~3;
  // Main loop: no k guards; loads unconditional from clamped bases, zeroed by cndmask.
  #pragma unroll 2
  for (int k0 = 0; k0 < K4; k0 += 4) {
    const int  ka = k0 + 2 * half;
    const long o0 = aBase + (long)ka * kA, o1 = o0 + kA;
    const long p0 = bBase + (long)ka * kB, p1 = p0 + kB;
    v2f ar, ai = {0.f, 0.f}, br, bi = {0.f, 0.f};
    {
      const float a0 = Ar[o0], a1 = Ar[o1];
      ar[0] = aOK ? a0 : 0.f;  ar[1] = aOK ? a1 : 0.f;
      const float b0 = Br[p0], b1 = Br[p1];
      br[0] = bOK ? b0 : 0.f;  br[1] = bOK ? b1 : 0.f;
      if (AC) { const float c0 = Ai[o0], c1 = Ai[o1]; ai[0] = aOK ? c0 : 0.f; ai[1] = aOK ? c1 : 0.f; }
      if (BC) { const float d0 = Bi[p0], d1 = Bi[p1]; bi[0] = bOK ? d0 : 0.f; bi[1] = bOK ? d1 : 0.f; }
    }
    crr = wmma4(ar, br, crr);
    if (BC)       cri = wmma4(ar, bi, cri);
    if (AC)       cir = wmma4(ai, br, cir);
    if (AC && BC) cii = wmma4(ai, bi, cii);
  }
  // K tail (0<rem<4): per-element k guards, clamped offsets.
  if (K4 < K) {
    const int  ka   = K4 + 2 * half;
    const bool k0ok = aOK && (ka < K);
    const bool k1ok = aOK && (ka + 1 < K);
    const bool q0ok = bOK && (ka < K);
    const bool q1ok = bOK && (ka + 1 < K);
    const long o0 = aBase + (k0ok ? (long)ka * kA : 0);
    const long o1 = aBase + (k1ok ? (long)(ka + 1) * kA : 0);
    const long p0 = bBase + (q0ok ? (long)ka * kB : 0);
    const long p1 = bBase + (q1ok ? (long)(ka + 1) * kB : 0);
    v2f ar, ai = {0.f, 0.f}, br, bi = {0.f, 0.f};
    {
      const float a0 = Ar[o0], a1 = Ar[o1];
      ar[0] = k0ok ? a0 : 0.f;  ar[1] = k1ok ? a1 : 0.f;
      const float b0 = Br[p0], b1 = Br[p1];
      br[0] = q0ok ? b0 : 0.f;  br[1] = q1ok ? b1 : 0.f;
      if (AC) { const float c0 = Ai[o0], c1 = Ai[o1]; ai[0] = k0ok ? c0 : 0.f; ai[1] = k1ok ? c1 : 0.f; }
      if (BC) { const float d0 = Bi[p0], d1 = Bi[p1]; bi[0] = q0ok ? d0 : 0.f; bi[1] = q1ok ? d1 : 0.f; }
    }
    crr = wmma4(ar, br, crr);
    if (BC)       cri = wmma4(ar, bi, cri);
    if (AC)       cir = wmma4(ai, br, cir);
    if (AC && BC) cii = wmma4(ai, bi, cii);
  }

  // (Ar+iAi)(Br + i*sB*Bi): Re = rr - sB*ii, Im = sB*ri + ir (zero accs cover real cases)
  const float sB = conjB ? -1.f : 1.f;
  const long  cz = z1 * zC1 + z2 * zC2;
  const int   cn = tn0 + l16;
  if (cn < N) {
    const int  n1 = cn / N2, n2 = cn - n1 * N2;
    const long cb = cz + (long)n1 * nC1 + (long)n2 * nC2;
    for (int v = 0; v < 8; ++v) {
      const int m = tm0 + v + 8 * half;   // C/D layout: VGPR v, lanes>=16 -> M+=8
      if (m < M) {
        const long off = cb + (long)m * mC;
        Cr[off] = crr[v] - sB * cii[v];
        if (CC) Ci[off] = sB * cri[v] + cir[v];
      }
    }
  }
}

// ---------------------------------------------------------------------------
// Constants: log-factorial table (fp64), SOFT quadrature weights, Wigner-d
// packed tensors, kernel bases, DFT twiddle tables.
// ---------------------------------------------------------------------------
__global__ void k_lgf_weights(double* lgf, float* w64, float* w20, float* w10) {
  const int t = threadIdx.x;
  if (t == 0) {
    double acc = 0.0; lgf[0] = 0.0;
    for (int n = 1; n < 64; ++n) { acc += log((double)n); lgf[n] = acc; }
  }
  if (t < 128) { // b = 64
    const int b = 64, j = t; double s = 0.0;
    for (int k = 0; k < b; ++k) s += sin((2*j+1)*(2*k+1)*PI_D/(4.0*b)) / (2*k+1);
    w64[j] = (float)(2.0/b * sin(PI_D*(2*j+1)/(4.0*b)) * s);
  }
  if (t < 40) { // b = 20
    const int b = 20, j = t; double s = 0.0;
    for (int k = 0; k < b; ++k) s += sin((2*j+1)*(2*k+1)*PI_D/(4.0*b)) / (2*k+1);
    w20[j] = (float)(2.0/b * sin(PI_D*(2*j+1)/(4.0*b)) * s);
  }
  if (t < 20) { // b = 10
    const int b = 10, j = t; double s = 0.0;
    for (int k = 0; k < b; ++k) s += sin((2*j+1)*(2*k+1)*PI_D/(4.0*b)) / (2*k+1);
    w10[j] = (float)(2.0/b * sin(PI_D*(2*j+1)/(4.0*b)) * s);
  }
}

__device__ double wig_d(int l, int mp, int mm, double cb, double sb, const double* lgf) {
  const double pref = 0.5 * (lgf[l+mp] + lgf[l-mp] + lgf[l+mm] + lgf[l-mm]);
  const int s0 = (mm - mp > 0) ? (mm - mp) : 0;
  const int s1 = ((l + mm) < (l - mp)) ? (l + mm) : (l - mp);
  double acc = 0.0;
  for (int s = s0; s <= s1; ++s) {
    const double ln = pref - (lgf[l+mm-s] + lgf[s] + lgf[mp-mm+s] + lgf[l-mp-s]);
    const double t = exp(ln) * pow(cb, (double)(2*l + mm - mp - 2*s)) * pow(sb, (double)(mp - mm + 2*s));
    acc += ((mp - mm + s) & 1) ? -t : t;
  }
  return acc;
}

// D[l][j][mi][ni] = d^l_{mv,nv}(beta_j) (* w_j*(2l+1)/2 if w); zero outside |mv|,|nv|<=l
__global__ void k_fill_so3(float* D, int nl, int nbeta, const float* w, const double* lgf) {
  const int nm = 2*nl - 1;
  const long total = (long)nl * nbeta * nm * nm;
  for (long idx = blockIdx.x * (long)blockDim.x + threadIdx.x; idx < total;
       idx += (long)gridDim.x * blockDim.x) {
    const int ni = (int)(idx % nm); long r = idx / nm;
    const int mi = (int)(r % nm);   r /= nm;
    const int j  = (int)(r % nbeta);
    const int l  = (int)(r / nbeta);
    const int mv = mi - (nl - 1), nv = ni - (nl - 1);
    float val = 0.f;
    if (mv >= -l && mv <= l && nv >= -l && nv <= l) {
      const double beta = PI_D * (2*j + 1) / (2.0 * nbeta); // soft_betas(b=nbeta/2)
      double v = wig_d(l, mv, nv, cos(0.5*beta), sin(0.5*beta), lgf);
      if (w) v *= (double)w[j] * (2*l + 1) * 0.5;
      val = (float)v;
    }
    D[idx] = val;
  }
}

// D[l][j][mi] = d^l_{mv,0}(beta_j) * w_j*(2l+1)/2
__global__ void k_fill_s2ana(float* D, int nl, int nbeta, const float* w, const double* lgf) {
  const int nm = 2*nl - 1;
  const long total = (long)nl * nbeta * nm;
  for (long idx = blockIdx.x * (long)blockDim.x + threadIdx.x; idx < total;
       idx += (long)gridDim.x * blockDim.x) {
    const int mi = (int)(idx % nm);
    const int j  = (int)((idx / nm) % nbeta);
    const int l  = (int)(idx / ((long)nm * nbeta));
    const int mv = mi - (nl - 1);
    float val = 0.f;
    if (mv >= -l && mv <= l) {
      const double beta = PI_D * (2*j + 1) / (2.0 * nbeta);
      val = (float)(wig_d(l, mv, 0, cos(0.5*beta), sin(0.5*beta), lgf)
                    * (double)w[j] * (2*l + 1) * 0.5);
    }
    D[idx] = val;
  }
}

// K[l][mi][p] = d^l_{mv,0}(pi/2) * e^{-i mv alpha_p}
__global__ void k_fill_s2_kbasis(float* Kr, float* Ki, int nl, int P, const double* lgf) {
  const int nm = 2*nl - 1;
  const long total = (long)nl * nm * P;
  const double r2 = 0.70710678118654752440; // cos(pi/4)
  for (long idx = blockIdx.x * (long)blockDim.x + threadIdx.x; idx < total;
       idx += (long)gridDim.x * blockDim.x) {
    const int p  = (int)(idx % P);
    const int mi = (int)((idx / P) % nm);
    const int l  = (int)(idx / ((long)P * nm));
    const int mv = mi - (nl - 1);
    float re = 0.f, im = 0.f;
    if (mv >= -l && mv <= l) {
      const double d = wig_d(l, mv, 0, r2, r2, lgf);
      const double a = 2.0 * PI_D * (double)(mv * p) / P;
      re = (float)(d * cos(a)); im = (float)(-d * sin(a));
    }
    Kr[idx] = re; Ki[idx] = im;
  }
}

// K[l][mi][ni][p] = d^l_{mv,nv}(pi/2) * e^{-i mv alpha_p}
__global__ void k_fill_so3_kbasis(float* Kr, float* Ki, int nl, int P, const double* lgf) {
  const int nm = 2*nl - 1;
  const long total = (long)nl * nm * nm * P;
  const double r2 = 0.70710678118654752440;
  for (long idx = blockIdx.x * (long)blockDim.x + threadIdx.x; idx < total;
       idx += (long)gridDim.x * blockDim.x) {
    const int p  = (int)(idx % P); long r = idx / P;
    const int ni = (int)(r % nm);  r /= nm;
    const int mi = (int)(r % nm);
    const int l  = (int)(r / nm);
    const int mv = mi - (nl - 1), nv = ni - (nl - 1);
    float re = 0.f, im = 0.f;
    if (mv >= -l && mv <= l && nv >= -l && nv <= l) {
      const double d = wig_d(l, mv, nv, r2, r2, lgf);
      const double a = 2.0 * PI_D * (double)(mv * p) / P;
      re = (float)(d * cos(a)); im = (float)(-d * sin(a));
    }
    Kr[idx] = re; Ki[idx] = im;
  }
}

// mMajor: T[mi][p], else T[p][mi]; value = scale * e^{i*sign*2pi*mv*p/npts}
__global__ void k_fill_twiddle(float* Tr, float* Ti, int npts, int nm, int nl,
                               float sign, float scale, int mMajor) {
  const long total = (long)npts * nm;
  for (long idx = blockIdx.x * (long)blockDim.x + threadIdx.x; idx < total;
       idx += (long)gridDim.x * blockDim.x) {
    int p, mi;
    if (mMajor) { p = (int)(idx % npts); mi = (int)(idx / npts); }
    else        { mi = (int)(idx % nm);  p  = (int)(idx / nm); }
    const int mv = mi - (nl - 1);
    const double th = (double)sign * 2.0 * PI_D * (double)(mv * p) / npts;
    Tr[idx] = scale * (float)cos(th);
    Ti[idx] = scale * (float)sin(th);
  }
}

// ---------------------------------------------------------------------------
// BN / pool / head
// ---------------------------------------------------------------------------
__global__ void k_bn_stats(const float* __restrict__ x, float* mv,
                           long chStride, long nOuter, long outerStride, long innerLen) {
  const int c = blockIdx.x;
  const long count = nOuter * innerLen;
  float s = 0.f, q = 0.f;
  for (long i = threadIdx.x; i < count; i += blockDim.x) {
    const long o = i / innerLen, in = i - o * innerLen;
    const float v = x[(long)c * chStride + o * outerStride + in];
    s += v; q += v * v;
  }
  __shared__ float sh[256], sq[256];
  sh[threadIdx.x] = s; sq[threadIdx.x] = q; __syncthreads();
  for (int st = 128; st > 0; st >>= 1) {
    if ((int)threadIdx.x < st) { sh[threadIdx.x] += sh[threadIdx.x + st]; sq[threadIdx.x] += sq[threadIdx.x + st]; }
    __syncthreads();
  }
  if (threadIdx.x == 0) {
    const float m = sh[0] / (float)count;
    mv[2*c] = m; mv[2*c + 1] = sq[0] / (float)count - m * m;
  }
}

// layout [...][ch][inner]: channel = (idx/inner) % nCh
__global__ void k_bn_relu(float* x, const float* mv, const float* g, const float* b,
                          int nCh, long innerLen, long total) {
  for (long idx = blockIdx.x * (long)blockDim.x + threadIdx.x; idx < total;
       idx += (long)gridDim.x * blockDim.x) {
    const int c = (int)((idx / innerLen) % nCh);
    const float y = (x[idx] - mv[2*c]) * rsqrtf(mv[2*c + 1] + 1e-5f) * g[c] + b[c];
    x[idx] = y > 0.f ? y : 0.f;
  }
}

// h layout [j][b][c][inner] -> out[b][c] = max over (j, inner)
__global__ void k_maxpool(const float* __restrict__ h, float* out,
                          int nJ, int nB, int nC, int inner) {
  const int b = blockIdx.x / nC, c = blockIdx.x % nC;
  float m = -3.402823e38f;
  const int tot = nJ * inner;
  for (int i = threadIdx.x; i < tot; i += blockDim.x) {
    const int j = i / inner, in = i - j * inner;
    m = fmaxf(m, h[((long)(j * nB + b) * nC + c) * inner + in]);
  }
  __shared__ float sh[256];
  sh[threadIdx.x] = m; __syncthreads();
  for (int st = 128; st > 0; st >>= 1) {
    if ((int)threadIdx.x < st) sh[threadIdx.x] = fmaxf(sh[threadIdx.x], sh[threadIdx.x + st]);
    __syncthreads();
  }
  if (threadIdx.x == 0) out[b * nC + c] = sh[0];
}

__global__ void k_std_feat(const float* pooled, float* z, int nB, int nF) {
  const int f = blockIdx.x * blockDim.x + threadIdx.x;
  if (f < nF) {
    float s = 0.f, q = 0.f;
    for (int b = 0; b < nB; ++b) { const float v = pooled[b * nF + f]; s += v; q += v * v; }
    const float m = s / nB;
    const float r = rsqrtf(q / nB - m * m + 1e-5f);
    for (int b = 0; b < nB; ++b) z[b * nF + f] = (pooled[b * nF + f] - m) * r;
  }
}

__global__ void k_head(const float* __restrict__ z, const float* __restrict__ W,
                       const float* __restrict__ bias, float* out, int nF, int nK) {
  const int b = blockIdx.x, t = threadIdx.x;
  __shared__ float lg[64];
  __shared__ float mx, ls;
  if (t < nK) {
    float s = bias[t];
    for (int f = 0; f < nF; ++f) s += z[b * nF + f] * W[t * nF + f];
    lg[t] = s;
  }
  __syncthreads();
  if (t == 0) {
    float m = lg[0];
    for (int k = 1; k < nK; ++k) m = fmaxf(m, lg[k]);
    float s = 0.f;
    for (int k = 0; k < nK; ++k) s += expf(lg[k] - m);
    mx = m; ls = logf(s);
  }
  __syncthreads();
  if (t < nK) out[b * nK + t] = lg[t] - mx - ls;
}

// ---------------------------------------------------------------------------
// Host orchestration
// ---------------------------------------------------------------------------
static inline void gemm(hipStream_t st,
    const float* Ar, const float* Ai, const float* Br, const float* Bi,
    float* Cr, float* Ci,
    int M, int N1, int N2, int K, int Z1, int Z2,
    long zA1, long zA2, long mA, long kA,
    long zB1, long zB2, long kB, long nB1, long nB2,
    long zC1, long zC2, long mC, long nC1, long nC2, int conjB)
{
  dim3 g((unsigned)((N1 * N2 + 31) / 32), (unsigned)((M + 31) / 32), (unsigned)(Z1 * Z2));
  const dim3 blk(128);
  const bool aC = (Ai != nullptr), bC = (Bi != nullptr), cC = (Ci != nullptr);
  if (aC && bC && cC)
    gemm_wmma_t<true, true, true><<<g, blk, 0, st>>>(Ar, Ai, Br, Bi, Cr, Ci,
        M, N1, N2, K, Z2, zA1, zA2, mA, kA, zB1, zB2, kB, nB1, nB2, zC1, zC2, mC, nC1, nC2, conjB);
  else if (aC && bC && !cC)
    gemm_wmma_t<true, true, false><<<g, blk, 0, st>>>(Ar, Ai, Br, Bi, Cr, Ci,
        M, N1, N2, K, Z2, zA1, zA2, mA, kA, zB1, zB2, kB, nB1, nB2, zC1, zC2, mC, nC1, nC2, conjB);
  else if (!aC && bC)
    gemm_wmma_t<false, true, true><<<g, blk, 0, st>>>(Ar, Ai, Br, Bi, Cr, Ci,
        M, N1, N2, K, Z2, zA1, zA2, mA, kA, zB1, zB2, kB, nB1, nB2, zC1, zC2, mC, nC1, nC2, conjB);
  else
    gemm_wmma_t<true, false, true><<<g, blk, 0, st>>>(Ar, Ai, Br, Bi, Cr, Ci,
        M, N1, N2, K, Z2, zA1, zA2, mA, kA, zB1, zB2, kB, nB1, nB2, zC1, zC2, mC, nC1, nC2, conjB);
}

static inline dim3 g1d(long total) { long b = (total + 255) / 256; if (b > 65535) b = 65535; return dim3((unsigned)b); }

extern "C" void kernel_launch(void* const* d_in, const int* in_sizes, int n_in,
                              void* d_out, int out_size, void* d_ws, size_t ws_size,
                              hipStream_t stream) {
  (void)in_sizes; (void)n_in; (void)out_size; (void)ws_size;
  const float* x    = (const float*)d_in[0];   // (32,6,128,128)
  const float* w1   = (const float*)d_in[1];   // (6,20,128)
  const float* g1   = (const float*)d_in[2];
  const float* b1   = (const float*)d_in[3];
  const float* w2   = (const float*)d_in[4];   // (20,60,40)
  const float* g2   = (const float*)d_in[5];
  const float* b2   = (const float*)d_in[6];
  const float* w3   = (const float*)d_in[7];   // (60,100,20)
  const float* g3   = (const float*)d_in[8];
  const float* b3   = (const float*)d_in[9];
  const float* outw = (const float*)d_in[10];  // (55,100)
  const float* outb = (const float*)d_in[11];  // (55,)
  float* out = (float*)d_out;                  // (32,55)
  float* ws  = (float*)d_ws;

  size_t off = 0;
  auto A = [&](size_t n) { size_t o = off; off += n; return o; };
  auto W = [&](size_t o) { return ws + o; };

  // ---- constants (persist through the call) ----
  const size_t oLGF  = A(128);                       // 64 doubles
  const size_t oW64  = A(128), oW20 = A(40), oW10 = A(24);
  const size_t oT1Ar = A(128*39),  oT1Ai = A(128*39);        // [a][mi], 1/128, sign -1
  const size_t oD1A  = A((size_t)20*128*39);                 // [l][j][mi]
  const size_t oK1r  = A((size_t)20*39*128), oK1i = A((size_t)20*39*128); // [l][mi][p]
  const size_t oD1S  = A((size_t)20*40*39*39);               // [l][j][mi][ni]
  const size_t oT1Sr = A(39*40),   oT1Si = A(39*40);         // [mi][p], sign +1
  const size_t oT2Ar = A(40*19),   oT2Ai = A(40*19);
  const size_t oD2A  = A((size_t)10*40*19*19);
  const size_t oK2r  = A((size_t)10*19*19*40), oK2i = A((size_t)10*19*19*40);
  const size_t oD2S  = A((size_t)10*20*19*19);
  const size_t oT2Sr = A(19*20),   oT2Si = A(19*20);
  const size_t oT3Ar = A(20*13),   oT3Ai = A(20*13);
  const size_t oD3A  = A((size_t)7*20*13*13);
  const size_t oK3r  = A((size_t)7*13*13*20), oK3i = A((size_t)7*13*13*20);
  const size_t oD3S  = A((size_t)7*14*13*13);
  const size_t oT3Sr = A(13*14),   oT3Si = A(13*14);
  const size_t oMV   = A(256);
  const size_t oPool = A(3200), oZf = A(3200);
  const size_t MARK1 = off;

  // ---- layer-1 scratch ----
  const size_t oXmr  = A(958464),   oXmi  = A(958464);    // [b,c,j][mi]
  const size_t oXH1r = A(149760),   oXH1i = A(149760);    // [l][b][mi][i]
  const size_t oYH1r = A(93600),    oYH1i = A(93600);     // [l][n][i][o]
  const size_t oZ1r  = A(19468800), oZ1i  = A(19468800);  // [mi][ni][l][b][o]
  const size_t oS1r  = A(38937600), oS1i  = A(38937600);  // [mi][ni][j][b][o]
  const size_t oT1r  = A(39936000), oT1i  = A(39936000);  // [mi][j,b,o][y]
  const size_t oH1   = MARK1;           // 40,960,000  (aliases dead Xm..Z1, fits)
  const size_t MARK2 = MARK1 + 41000000;

  // ---- layer-2 scratch (reuses layer-1 peak region) ----
  off = MARK2;
  const size_t oQ2r  = A(19456000), oQ2i  = A(19456000);
  const size_t oX2r  = A(9241600),  oX2i  = A(9241600);
  const size_t oXH2r = A(2310400),  oXH2i = A(2310400);
  const size_t oYH2r = A(4332000),  oYH2i = A(4332000);
  const size_t oZ2r  = A(6931200),  oZ2i  = A(6931200);
  const size_t oS2r  = A(13862400), oS2i  = A(13862400);
  const size_t oT2r  = A(14592000), oT2i  = A(14592000);
  const size_t oH2   = MARK1;           // 15,360,000 (h1 dead by then)

  // ---- layer-3 scratch (reuses layer-2 scratch) ----
  off = MARK2;
  const size_t oQ3r  = A(9984000), oQ3i  = A(9984000);
  const size_t oX3r  = A(6489600), oX3i  = A(6489600);
  const size_t oXH3r = A(2271360), oXH3i = A(2271360);
  const size_t oYH3r = A(7098000), oYH3i = A(7098000);
  const size_t oZ3r  = A(3785600), oZ3i  = A(3785600);
  const size_t oS3r  = A(7571200), oS3i  = A(7571200);
  const size_t oT3r  = A(8153600), oT3i  = A(8153600);
  const size_t oH3   = A(8780800);

  double* lgf = (double*)W(oLGF);

  // ================= constants =================
  k_lgf_weights<<<dim3(1), dim3(128), 0, stream>>>(lgf, W(oW64), W(oW20), W(oW10));
  k_fill_twiddle<<<g1d(128*39), 256, 0, stream>>>(W(oT1Ar), W(oT1Ai), 128, 39, 20, -1.f, 1.f/128.f, 0);
  k_fill_twiddle<<<g1d(40*19),  256, 0, stream>>>(W(oT2Ar), W(oT2Ai), 40, 19, 10, -1.f, 1.f/40.f, 0);
  k_fill_twiddle<<<g1d(20*13),  256, 0, stream>>>(W(oT3Ar), W(oT3Ai), 20, 13, 7,  -1.f, 1.f/20.f, 0);
  k_fill_twiddle<<<g1d(39*40),  256, 0, stream>>>(W(oT1Sr), W(oT1Si), 40, 39, 20,  1.f, 1.f, 1);
  k_fill_twiddle<<<g1d(19*20),  256, 0, stream>>>(W(oT2Sr), W(oT2Si), 20, 19, 10,  1.f, 1.f, 1);
  k_fill_twiddle<<<g1d(13*14),  256, 0, stream>>>(W(oT3Sr), W(oT3Si), 14, 13, 7,   1.f, 1.f, 1);
  k_fill_s2ana<<<g1d(20*128*39), 256, 0, stream>>>(W(oD1A), 20, 128, W(oW64), lgf);
  k_fill_so3<<<g1d((long)20*40*39*39), 256, 0, stream>>>(W(oD1S), 20, 40, nullptr, lgf);
  k_fill_so3<<<g1d((long)10*40*19*19), 256, 0, stream>>>(W(oD2A), 10, 40, W(oW20), lgf);
  k_fill_so3<<<g1d((long)10*20*19*19), 256, 0, stream>>>(W(oD2S), 10, 20, nullptr, lgf);
  k_fill_so3<<<g1d((long)7*20*13*13),  256, 0, stream>>>(W(oD3A), 7, 20, W(oW10), lgf);
  k_fill_so3<<<g1d((long)7*14*13*13),  256, 0, stream>>>(W(oD3S), 7, 14, nullptr, lgf);
  k_fill_s2_kbasis<<<g1d((long)20*39*128), 256, 0, stream>>>(W(oK1r), W(oK1i), 20, 128, lgf);
  k_fill_so3_kbasis<<<g1d((long)10*19*19*40), 256, 0, stream>>>(W(oK2r), W(oK2i), 10, 40, lgf);
  k_fill_so3_kbasis<<<g1d((long)7*13*13*20),  256, 0, stream>>>(W(oK3r), W(oK3i), 7, 20, lgf);

  // ================= layer 1: S2 conv (b 64 -> 20) =================
  // Xm[b,c,j][mi] = sum_a x * T1a           (A real, B cplx)
  gemm(stream, x, nullptr, W(oT1Ar), W(oT1Ai), W(oXmr), W(oXmi),
       24576, 39, 1, 128, 1, 1,  0,0,128,1,  0,0,39,1,0,  0,0,39,1,0, 0);
  // xh1[l][b][mi][i] = sum_j d1a * Xm       (batch mi=39)
  gemm(stream, W(oD1A), nullptr, W(oXmr), W(oXmi), W(oXH1r), W(oXH1i),
       20, 32, 6, 128, 39, 1,  1,0,4992,39,  1,0,39,29952,4992,  6,0,7488,234,1, 0);
  // yh1[l][n][i][o] = sum_p K1 * w1         (batch l=20; A cplx, B real)
  gemm(stream, W(oK1r), W(oK1i), w1, nullptr, W(oYH1r), W(oYH1i),
       39, 6, 20, 128, 20, 1,  4992,0,128,1,  0,0,1,2560,128,  4680,0,120,20,1, 0);
  // Z1[mi][ni][l][b][o] = sum_i xh1 * conj(yh1)   (batch (l,b))
  gemm(stream, W(oXH1r), W(oXH1i), W(oYH1r), W(oYH1i), W(oZ1r), W(oZ1i),
       39, 39, 20, 6, 20, 32,  7488,234,6,1,  4680,0,20,120,1,  640,20,499200,12800,1, 1);
  // S1[mi][ni][j][b][o] = sum_l d1s * Z1    (batch (mi,ni))
  gemm(stream, W(oD1S), nullptr, W(oZ1r), W(oZ1i), W(oS1r), W(oS1i),
       40, 32, 20, 20, 39, 39,  39,1,1521,60840,  499200,12800,640,20,1,  998400,25600,640,20,1, 0);
  // T1[mi][r=(j,b,o)][y] = sum_ni S1 * T1s  (batch mi)
  gemm(stream, W(oS1r), W(oS1i), W(oT1Sr), W(oT1Si), W(oT1r), W(oT1i),
       25600, 40, 1, 39, 39, 1,  998400,0,1,25600,  0,0,40,1,0,  1024000,0,40,1,0, 0);
  // h1[r][x][y] = Re sum_mi T1s[mi][x] * T1[mi][r][y]   (batch r; real out)
  gemm(stream, W(oT1Sr), W(oT1Si), W(oT1r), W(oT1i), W(oH1), nullptr,
       40, 40, 1, 39, 25600, 1,  0,0,1,40,  40,0,1024000,1,0,  1600,0,40,1,0, 0);
  // BN1 + ReLU (h1 layout [j][b][o][x*y], ch=o)
  k_bn_stats<<<dim3(20), dim3(256), 0, stream>>>(W(oH1), W(oMV), 1600, 1280, 32000, 1600);
  k_bn_relu<<<g1d(40960000), 256, 0, stream>>>(W(oH1), W(oMV), g1, b1, 20, 1600, 40960000L);

  // ================= layer 2: SO3 conv (b 20 -> 10) =================
  gemm(stream, W(oH1), nullptr, W(oT2Ar), W(oT2Ai), W(oQ2r), W(oQ2i),
       1024000, 19, 1, 40, 1, 1,  0,0,40,1,  0,0,19,1,0,  0,0,19,1,0, 0);
  gemm(stream, W(oT2Ar), W(oT2Ai), W(oQ2r), W(oQ2i), W(oX2r), W(oX2i),
       19, 19, 1, 40, 25600, 1,  0,0,1,19,  760,0,19,1,0,  1,0,486400,25600,0, 0);
  gemm(stream, W(oD2A), nullptr, W(oX2r), W(oX2i), W(oXH2r), W(oXH2i),
       10, 32, 20, 40, 19, 19,  19,1,14440,361,  486400,25600,640,20,1,  380,20,231040,7220,1, 0);
  gemm(stream, W(oK2r), W(oK2i), w2, nullptr, W(oYH2r), W(oYH2i),
       361, 20, 60, 40, 10, 1,  14440,0,40,1,  0,0,1,2400,40,  433200,0,1200,60,1, 0);
  gemm(stream, W(oXH2r), W(oXH2i), W(oYH2r), W(oYH2i), W(oZ2r), W(oZ2i),
       19, 19, 60, 380, 10, 32,  231040,7220,380,1,  433200,0,60,22800,1,  1920,60,364800,19200,1, 1);
  gemm(stream, W(oD2S), nullptr, W(oZ2r), W(oZ2i), W(oS2r), W(oS2i),
       20, 32, 60, 10, 19, 19,  19,1,361,7220,  364800,19200,1920,60,1,  729600,38400,1920,60,1, 0);
  gemm(stream, W(oS2r), W(oS2i), W(oT2Sr), W(oT2Si), W(oT2r), W(oT2i),
       38400, 20, 1, 19, 19, 1,  729600,0,1,38400,  0,0,20,1,0,  768000,0,20,1,0, 0);
  gemm(stream, W(oT2Sr), W(oT2Si), W(oT2r), W(oT2i), W(oH2), nullptr,
       20, 20, 1, 19, 38400, 1,  0,0,1,20,  20,0,768000,1,0,  400,0,20,1,0, 0);
  k_bn_stats<<<dim3(60), dim3(256), 0, stream>>>(W(oH2), W(oMV), 400, 640, 24000, 400);
  k_bn_relu<<<g1d(15360000), 256, 0, stream>>>(W(oH2), W(oMV), g2, b2, 60, 400, 15360000L);

  // ================= layer 3: SO3 conv (b 10 -> 7) =================
  gemm(stream, W(oH2), nullptr, W(oT3Ar), W(oT3Ai), W(oQ3r), W(oQ3i),
       768000, 13, 1, 20, 1, 1,  0,0,20,1,  0,0,13,1,0,  0,0,13,1,0, 0);
  gemm(stream, W(oT3Ar), W(oT3Ai), W(oQ3r), W(oQ3i), W(oX3r), W(oX3i),
       13, 13, 1, 20, 38400, 1,  0,0,1,13,  260,0,13,1,0,  1,0,499200,38400,0, 0);
  gemm(stream, W(oD3A), nullptr, W(oX3r), W(oX3i), W(oXH3r), W(oXH3i),
       7, 32, 60, 20, 13, 13,  13,1,3380,169,  499200,38400,1920,60,1,  780,60,324480,10140,1, 0);
  gemm(stream, W(oK3r), W(oK3i), w3, nullptr, W(oYH3r), W(oYH3i),
       169, 60, 100, 20, 7, 1,  3380,0,20,1,  0,0,1,2000,20,  1014000,0,6000,100,1, 0);
  gemm(stream, W(oXH3r), W(oXH3i), W(oYH3r), W(oYH3i), W(oZ3r), W(oZ3i),
       13, 13, 100, 780, 7, 32,  324480,10140,780,1,  1014000,0,100,78000,1,  3200,100,291200,22400,1, 1);
  gemm(stream, W(oD3S), nullptr, W(oZ3r), W(oZ3i), W(oS3r), W(oS3i),
       14, 32, 100, 7, 13, 13,  13,1,169,2366,  291200,22400,3200,100,1,  582400,44800,3200,100,1, 0);
  gemm(stream, W(oS3r), W(oS3i), W(oT3Sr), W(oT3Si), W(oT3r), W(oT3i),
       44800, 14, 1, 13, 13, 1,  582400,0,1,44800,  0,0,14,1,0,  627200,0,14,1,0, 0);
  gemm(stream, W(oT3Sr), W(oT3Si), W(oT3r), W(oT3i), W(oH3), nullptr,
       14, 14, 1, 13, 44800, 1,  0,0,1,14,  14,0,627200,1,0,  196,0,14,1,0, 0);
  k_bn_stats<<<dim3(100), dim3(256), 0, stream>>>(W(oH3), W(oMV), 196, 448, 19600, 196);
  k_bn_relu<<<g1d(8780800), 256, 0, stream>>>(W(oH3), W(oMV), g3, b3, 100, 196, 8780800L);

  // ================= head =================
  k_maxpool<<<dim3(3200), dim3(256), 0, stream>>>(W(oH3), W(oPool), 14, 32, 100, 196);
  k_std_feat<<<dim3(1), dim3(128), 0, stream>>>(W(oPool), W(oZf), 32, 100);
  k_head<<<dim3(32), dim3(64), 0, stream>>>(W(oZf), outw, outb, out, 100, 55);
}